// EuclideanPairwiseDistances_23081154248838
// MI455X (gfx1250) — compile-verified
//
#include <hip/hip_runtime.h>
#include <hip/hip_bf16.h>

typedef __attribute__((ext_vector_type(2))) float v2f;
typedef __attribute__((ext_vector_type(8))) float v8f;

#define B_    4
#define H_    256
#define L_    1024
#define N_    128
#define NPAIR 8128          // 128*127/2
#define XSTR  258           // 256 + 2 pad -> conflict-free LDS b64 row access
#define EPS_  1e-8f

// ---------------------------------------------------------------------------
// Kernel 1: masked mean over L (skipping l=0).  Output X[b][h][n]  (B,H,N).
// grid = B*H blocks, 256 threads (8 waves).  Wave w handles l = 1+w, step 8.
// Lane loads float4 over n -> 512B coalesced per wave per row.  Mask bytes
// (jnp bool_ = 1 byte) loaded as uchar4; count accumulated in f32 (<=1023,
// exact).  Cross-wave combine via LDS, single b128 store per lane.
// ---------------------------------------------------------------------------
__global__ void __launch_bounds__(256) masked_mean_kernel(
    const float* __restrict__ inp, const unsigned char* __restrict__ mask,
    float* __restrict__ X)
{
    const int bh   = blockIdx.x;         // b*H + h
    const int b    = bh >> 8;            // H = 256
    const int t    = threadIdx.x;
    const int lane = t & 31;
    const int lc   = t >> 5;             // l-chunk 0..7 (one wave each)
    const int n    = lane << 2;          // n base: 0..124

    const float*         ip = inp  + (size_t)bh * L_ * N_;
    const unsigned char* mp = mask + (size_t)b  * L_ * N_;

    float4 acc = make_float4(0.f, 0.f, 0.f, 0.f);
    float4 cnt = make_float4(0.f, 0.f, 0.f, 0.f);
    for (int l = 1 + lc; l < L_; l += 8) {
        const float4 v  = *(const float4*)(ip + (size_t)l * N_ + n);
        const uchar4 m4 = *(const uchar4*)(mp + (size_t)l * N_ + n);
        acc.x += m4.x ? v.x : 0.f;  cnt.x += m4.x ? 1.f : 0.f;
        acc.y += m4.y ? v.y : 0.f;  cnt.y += m4.y ? 1.f : 0.f;
        acc.z += m4.z ? v.z : 0.f;  cnt.z += m4.z ? 1.f : 0.f;
        acc.w += m4.w ? v.w : 0.f;  cnt.w += m4.w ? 1.f : 0.f;
    }

    __shared__ float4 sa[256];
    __shared__ float4 sc[256];
    sa[t] = acc;  sc[t] = cnt;
    __syncthreads();

    if (lc == 0) {
        for (int c = 1; c < 8; ++c) {
            const float4 a2 = sa[lane + 32 * c];
            const float4 c2 = sc[lane + 32 * c];
            acc.x += a2.x; acc.y += a2.y; acc.z += a2.z; acc.w += a2.w;
            cnt.x += c2.x; cnt.y += c2.y; cnt.z += c2.z; cnt.w += c2.w;
        }
        float4 r;
        r.x = acc.x / fmaxf(cnt.x, 1.f);
        r.y = acc.y / fmaxf(cnt.y, 1.f);
        r.z = acc.z / fmaxf(cnt.z, 1.f);
        r.w = acc.w / fmaxf(cnt.w, 1.f);
        *(float4*)(X + (size_t)bh * N_ + n) = r;
    }
}

// ---------------------------------------------------------------------------
// Kernel 2: per-batch pairwise distances via WMMA Gram matrix.
// grid = B blocks, 512 threads (16 waves).  X (B,H,N) is loaded linearly
// (coalesced) and transposed into LDS Xs[n*XSTR + h].  Row norms in LDS.
// 36 lower-triangle 16x16 tiles; per tile a K=256 loop of
// V_WMMA_F32_16X16X4_F32 (full f32 precision).
// A 16x4 f32 layout (ISA 7.12.2): lane<16 -> M=lane, K={0,1}; lane>=16 ->
// M=lane-16, K={2,3}.  B 4x16 mirrors it with M<->N, so both fragments are
// an 8B LDS load of two consecutive K values from row (base + lane&15).
// ---------------------------------------------------------------------------
__global__ void __launch_bounds__(512) pairwise_dist_kernel(
    const float* __restrict__ X, float* __restrict__ out)
{
    __shared__ float Xs[N_ * XSTR];      // 132,096 B (<= 320 KB WGP LDS)
    __shared__ float norms[N_];

    const int b = blockIdx.x;
    const int t = threadIdx.x;
    const float* Xb = X + (size_t)b * H_ * N_;

    // transpose-load: linear global (coalesced), scattered-but-conflict-free LDS
    for (int e = t; e < H_ * N_; e += 512) {
        const int n = e & (N_ - 1);
        const int h = e >> 7;            // N_ = 128
        Xs[n * XSTR + h] = Xb[e];
    }
    __syncthreads();

    if (t < N_) {
        float s = 0.f;
        for (int h = 0; h < H_; ++h) {
            const float v = Xs[t * XSTR + h];
            s += v * v;
        }
        norms[t] = s;
    }
    __syncthreads();

    const int wave = t >> 5;
    const int lane = t & 31;
    const int half = lane >> 4;          // 0: K 0,1 | 1: K 2,3
    const int lr   = lane & 15;

    for (int tidx = wave; tidx < 36; tidx += 16) {
        // decode tile index -> (ti, tj), ti >= tj, row-major over ti
        int ti = 0, rem = tidx;
        while (rem > ti) { rem -= ti + 1; ++ti; }
        const int tj = rem;
        const int i0 = ti * 16, j0 = tj * 16;

        const float* arow = &Xs[(i0 + lr) * XSTR];
        const float* brow = &Xs[(j0 + lr) * XSTR];

        v8f acc = {0.f, 0.f, 0.f, 0.f, 0.f, 0.f, 0.f, 0.f};
#if __has_builtin(__builtin_amdgcn_wmma_f32_16x16x4_f32)
#pragma unroll 4
        for (int k = 0; k < H_; k += 4) {
            const int ka = k + 2 * half;
            const v2f a  = *(const v2f*)(arow + ka);
            const v2f bb = *(const v2f*)(brow + ka);
            acc = __builtin_amdgcn_wmma_f32_16x16x4_f32(
                false, a, false, bb, (short)0, acc, false, false);
        }
#else
        // fallback: scalar dot products in the same C-layout positions
        for (int v = 0; v < 8; ++v) {
            const int i = i0 + v + 8 * half;
            const int j = j0 + lr;
            float s = 0.f;
            for (int k = 0; k < H_; ++k)
                s += Xs[i * XSTR + k] * Xs[j * XSTR + k];
            acc[v] = s;
        }
#endif
        // C/D layout: VGPR v, lane -> M = v + 8*half (row i), N = lr (col j)
        const int   j  = j0 + lr;
        const float nj = norms[j];
#pragma unroll
        for (int v = 0; v < 8; ++v) {
            const int i = i0 + v + 8 * half;
            if (i > j) {
                float d2 = norms[i] + nj - 2.f * acc[v];
                d2 = fmaxf(d2, 0.f) + EPS_;
                out[(size_t)b * NPAIR + ((i * (i - 1)) >> 1) + j] = sqrtf(d2);
            }
        }
    }
}

extern "C" void kernel_launch(void* const* d_in, const int* in_sizes, int n_in,
                              void* d_out, int out_size, void* d_ws, size_t ws_size,
                              hipStream_t stream) {
    const float*         inp  = (const float*)d_in[0];
    const unsigned char* mask = (const unsigned char*)d_in[1];  // jnp bool_ = 1B
    float*               out  = (float*)d_out;
    float*               X    = (float*)d_ws;                   // B*H*N f32 = 512 KB

    masked_mean_kernel<<<B_ * H_, 256, 0, stream>>>(inp, mask, X);
    pairwise_dist_kernel<<<B_, 512, 0, stream>>>(X, out);
}